// SmoothingLayer_2293512536845
// MI455X (gfx1250) — compile-verified
//
#include <hip/hip_runtime.h>
#include <stdint.h>

// Problem constants (fixed by the reference): B=8, NUM_CHUNKS=1024, D_MODEL=512, L=8192
#define BB 8
#define CC 1024
#define DD 512
#define LL 8192
#define NSEG 64
#define SEGLEN (LL / NSEG)   // 128

// ---- workspace layout (total ~2.3 MB) ----
#define OFS_IDX   ((size_t)0)                                   // int32 [B][L]       = 256 KB
#define OFS_M     (OFS_IDX + (size_t)BB * LL * 4)               // f32   [B][NSEG]    (padded)
#define OFS_CP    (OFS_M + 4096)                                // f32   [B][NSEG][D] = 1 MB
#define OFS_CARRY (OFS_CP + (size_t)BB * NSEG * DD * 4)         // f32   [B][NSEG][D] = 1 MB

// ---------- gfx1250 async copy: global -> LDS, 4B per lane (ASYNCcnt-tracked) ----------
__device__ __forceinline__ void async_copy_b32_to_lds(const void* gptr, void* lptr) {
  // Low 32 bits of a generic shared-aperture address are the LDS byte offset.
  unsigned lds = (unsigned)(uintptr_t)lptr;
  asm volatile("global_load_async_to_lds_b32 %0, %1, off"
               :: "v"(lds), "v"(gptr) : "memory");
}
__device__ __forceinline__ void wait_asynccnt0() {
  asm volatile("s_wait_asynccnt 0" ::: "memory");
}

// ---------- K1: boundary cumsum -> chunk index per position (-1 == invalid) ----------
__global__ void k_boundaries(const float* __restrict__ probs, int* __restrict__ idxout) {
  const int b   = blockIdx.x;       // 8 blocks
  const int tid = threadIdx.x;      // 1024 threads, each owns 8 consecutive t
  __shared__ int s_sum[1024];

  const int base = tid * 8;
  int loc[8];
  int run = 0;
#pragma unroll
  for (int u = 0; u < 8; ++u) {
    float p = probs[(size_t)b * LL + base + u];
    run += (p > 0.5f) ? 1 : 0;
    loc[u] = run;                   // inclusive within the 8-elem strip
  }
  s_sum[tid] = run;
  __syncthreads();

  // Hillis-Steele inclusive scan over 1024 partials
  for (int off = 1; off < 1024; off <<= 1) {
    int add = (tid >= off) ? s_sum[tid - off] : 0;
    __syncthreads();
    s_sum[tid] += add;
    __syncthreads();
  }
  const int excl  = (tid == 0) ? 0 : s_sum[tid - 1];
  const int total = s_sum[1023];    // any(mask) over the row

#pragma unroll
  for (int u = 0; u < 8; ++u) {
    const int t   = base + u;
    const int cnt = excl + loc[u];  // inclusive boundary count at t
    int o;
    if (total == 0) {
      o = t >> 3;                   // uniform fallback: step = L/NUM_CHUNKS = 8, always valid
    } else {
      int id = cnt - 1;
      if (id < 0) id = 0;
      if (id > CC - 1) id = CC - 1;
      o = (cnt >= 1 && cnt <= CC) ? id : -1;
    }
    idxout[(size_t)b * LL + t] = o;
  }
}

// ---------- K2: per-segment product M = prod(1 - a_eff), a_eff[0] = 1 ----------
__global__ void k_segprod(const float* __restrict__ probs, float* __restrict__ Mout) {
  const int j   = threadIdx.x;      // B*NSEG = 512 threads, 1 block
  const int b   = j / NSEG;
  const int seg = j % NSEG;
  float p = 1.0f;
  for (int k = 0; k < SEGLEN; ++k) {
    const int t = seg * SEGLEN + k;
    const float a = (t == 0) ? 1.0f : probs[(size_t)b * LL + t];
    p *= (1.0f - a);
  }
  Mout[b * NSEG + seg] = p;
}

// ---------- K3 (FINAL=false): segment partials C;  K5 (FINAL=true): final scan + store ----------
template <bool FINAL>
__global__ void k_scan(const float* __restrict__ X, const float* __restrict__ probs,
                       const int* __restrict__ idxb, const float* __restrict__ carry,
                       float* __restrict__ Cp, float* __restrict__ out) {
  const int seg = blockIdx.x;       // 0..NSEG-1
  const int b   = blockIdx.y;       // 0..B-1
  const int d   = threadIdx.x;      // 0..511
  const int tid = threadIdx.x;

  __shared__ float s_a[SEGLEN];
  __shared__ int   s_idx[SEGLEN];

  // Stage wave-uniform per-t metadata via gfx1250 async global->LDS copies:
  // lanes 0..SEGLEN-1 fetch a[], lanes SEGLEN..2*SEGLEN-1 fetch idx[]; one instruction.
  if (tid < 2 * SEGLEN) {
    const bool lo = tid < SEGLEN;
    const size_t row = (size_t)b * LL + (size_t)seg * SEGLEN;
    const void* g = lo ? (const void*)(probs + row + tid)
                       : (const void*)(idxb + row + (tid - SEGLEN));
    void* l = lo ? (void*)&s_a[tid] : (void*)&s_idx[tid - SEGLEN];
    async_copy_b32_to_lds(g, l);
  }
  wait_asynccnt0();
  __syncthreads();

  float s;
  if constexpr (FINAL) s = carry[((size_t)b * NSEG + seg) * DD + d];
  else                 s = 0.0f;

  const float* Xb = X + (size_t)b * CC * DD + d;
  float* ob = nullptr;
  if constexpr (FINAL)
    ob = out + ((size_t)b * LL + (size_t)seg * SEGLEN) * DD + d;

  for (int k0 = 0; k0 < SEGLEN; k0 += 8) {
    // pull the next chunk's first gather row toward the WGP (global_prefetch_b8)
    if (k0 + 8 < SEGLEN) {
      const int idn = s_idx[k0 + 8];
      if (idn >= 0) __builtin_prefetch(Xb + (size_t)idn * DD, 0, 1);
    }
    float ev[8];
#pragma unroll
    for (int u = 0; u < 8; ++u) {       // batch 8 independent gathers (load clause; L2-resident X)
      const int id = s_idx[k0 + u];
      ev[u] = (id >= 0) ? Xb[(size_t)id * DD] : 0.0f;
    }
#pragma unroll
    for (int u = 0; u < 8; ++u) {       // dependent 2-op chain per step
      float af = s_a[k0 + u];
      if (seg == 0 && (k0 + u) == 0) af = 1.0f;   // encodes s[0] = e[0]
      s = fmaf(af, ev[u] - s, s);        // s = a*e + (1-a)*s
      if constexpr (FINAL)
        __builtin_nontemporal_store(s, &ob[(size_t)(k0 + u) * DD]);  // NT: don't evict X from L2
    }
  }
  if constexpr (!FINAL)
    __builtin_nontemporal_store(s, &Cp[((size_t)b * NSEG + seg) * DD + d]);
}

// ---------- K4: sequential composition of NSEG segment carries per (b, d) ----------
__global__ void k_carry(const float* __restrict__ Mv, const float* __restrict__ Cp,
                        float* __restrict__ carry) {
  const int b = blockIdx.x;
  const int d = threadIdx.x;
  float c = 0.0f;
  for (int seg = 0; seg < NSEG; ++seg) {
    __builtin_nontemporal_store(c, &carry[((size_t)b * NSEG + seg) * DD + d]);
    c = Mv[b * NSEG + seg] * c + Cp[((size_t)b * NSEG + seg) * DD + d];
  }
}

extern "C" void kernel_launch(void* const* d_in, const int* in_sizes, int n_in,
                              void* d_out, int out_size, void* d_ws, size_t ws_size,
                              hipStream_t stream) {
  const float* X     = (const float*)d_in[0];   // (8, 1024, 512) f32
  const float* probs = (const float*)d_in[1];   // (8, 8192) f32
  // d_in[2] = original_length (always 8192) — baked in as LL.

  char*  ws    = (char*)d_ws;                   // needs ~2.3 MB
  int*   idxb  = (int*)  (ws + OFS_IDX);
  float* Mv    = (float*)(ws + OFS_M);
  float* Cp    = (float*)(ws + OFS_CP);
  float* carry = (float*)(ws + OFS_CARRY);
  float* out   = (float*)d_out;                 // (8, 8192, 512) f32

  k_boundaries<<<BB, 1024, 0, stream>>>(probs, idxb);
  k_segprod  <<<1, BB * NSEG, 0, stream>>>(probs, Mv);
  k_scan<false><<<dim3(NSEG, BB), DD, 0, stream>>>(X, probs, idxb, nullptr, Cp, nullptr);
  k_carry    <<<BB, DD, 0, stream>>>(Mv, Cp, carry);
  k_scan<true> <<<dim3(NSEG, BB), DD, 0, stream>>>(X, probs, idxb, carry, nullptr, out);
}